// HopGatedGATv2Conv_9285719294404
// MI455X (gfx1250) — compile-verified
//
#include <hip/hip_runtime.h>
#include <hip/hip_bf16.h>
#include <math.h>

#define IN_DIM   256
#define HC       256      // H * C
#define H_HEADS  4
#define C_DIM    64
#define HOPS     3
#define NEG_SLOPE 0.2f

typedef __attribute__((ext_vector_type(2))) float v2f;
typedef __attribute__((ext_vector_type(8))) float v8f;

// ---------------------------------------------------------------------------
// Projection GEMM: Y[n, 0:256] = X[n, 0:256] @ W[256, 256] + b  (row-major)
// One wave computes a 16x32 output tile (two accumulators) with
// V_WMMA_F32_16X16X4_F32; each LDS A-frag feeds two WMMAs.
// Block = 128 threads (4 waves). All waves share the M-tile A panel in LDS,
// each wave owns a 32-wide N-slice. Grid = (ceil(N/16), 2).
// A-frag layout (ISA 7.12.2, 32-bit A 16x4): lane<16 -> M=lane, K={k,k+1};
// lane>=16 -> M=lane-16, K={k+2,k+3}. C/D: VGPR v -> M=v (+8 for hi half),
// N = lane&15.
// ---------------------------------------------------------------------------
__global__ __launch_bounds__(128)
void proj_gemm_wmma(const float* __restrict__ X, const float* __restrict__ W,
                    const float* __restrict__ bvec, float* __restrict__ Y,
                    int nrows)
{
    __shared__ __align__(16) float As[16 * IN_DIM];   // 16 KB A panel
    const int m0   = blockIdx.x * 16;
    const int wave = threadIdx.x >> 5;
    const int lane = threadIdx.x & 31;
    const int n0   = (blockIdx.y * 8 + wave * 2) * 16;   // 32 cols per wave

    const bool full = (m0 + 16 <= nrows);
    if (full) {
        // vectorized stage: 1024 float4, 8 per thread -> global_load_b128
        const float4* Xv = (const float4*)(X + (size_t)m0 * IN_DIM);
        float4* Av = (float4*)As;
        for (int i = threadIdx.x; i < 16 * IN_DIM / 4; i += 128)
            Av[i] = Xv[i];
    } else {
        for (int i = threadIdx.x; i < 16 * IN_DIM; i += 128) {
            int r = i >> 8, c = i & 255;
            int gr = m0 + r;
            As[i] = (gr < nrows) ? X[(size_t)gr * IN_DIM + c] : 0.0f;
        }
    }
    __syncthreads();

    v8f acc0 = {}, acc1 = {};
    const int mA = lane & 15;
    const int kk = (lane >> 4) * 2;      // 0 or 2
    const int nB = lane & 15;

    for (int k = 0; k < IN_DIM; k += 4) {
        v2f a, b0, b1;
        a[0] = As[mA * IN_DIM + k + kk];
        a[1] = As[mA * IN_DIM + k + kk + 1];
        const float* w0 = W + (size_t)(k + kk) * HC + n0 + nB;
        b0[0] = w0[0];   b0[1] = w0[HC];
        b1[0] = w0[16];  b1[1] = w0[HC + 16];
        acc0 = __builtin_amdgcn_wmma_f32_16x16x4_f32(
            false, a, false, b0, (short)0, acc0, false, false);
        acc1 = __builtin_amdgcn_wmma_f32_16x16x4_f32(
            false, a, false, b1, (short)0, acc1, false, false);
    }

    const int col   = n0 + (lane & 15);
    const float bb0 = bvec[col];
    const float bb1 = bvec[col + 16];
    const int rbase = m0 + ((lane >> 4) * 8);
    if (full) {
#pragma unroll
        for (int v = 0; v < 8; ++v) {
            int row = rbase + v;
            Y[(size_t)row * HC + col]      = acc0[v] + bb0;
            Y[(size_t)row * HC + col + 16] = acc1[v] + bb1;
        }
    } else {
#pragma unroll
        for (int v = 0; v < 8; ++v) {
            int row = rbase + v;
            if (row < nrows) {
                Y[(size_t)row * HC + col]      = acc0[v] + bb0;
                Y[(size_t)row * HC + col + 16] = acc1[v] + bb1;
            }
        }
    }
}

// ---------------------------------------------------------------------------
// Gate: logits = x @ W_gate + b_gate ; softmax over HOPS. One wave per node.
// ---------------------------------------------------------------------------
__global__ __launch_bounds__(256)
void gate_kernel(const float* __restrict__ x, const float* __restrict__ Wg,
                 const float* __restrict__ bg, float* __restrict__ gw, int n_nodes)
{
    int n = blockIdx.x * 8 + (threadIdx.x >> 5);
    if (n >= n_nodes) return;
    int lane = threadIdx.x & 31;
    const float* px = x + (size_t)n * IN_DIM;
    float a0 = 0.f, a1 = 0.f, a2 = 0.f;
    for (int i = lane; i < IN_DIM; i += 32) {
        float xv = px[i];
        a0 += xv * Wg[i * HOPS + 0];
        a1 += xv * Wg[i * HOPS + 1];
        a2 += xv * Wg[i * HOPS + 2];
    }
#pragma unroll
    for (int off = 16; off > 0; off >>= 1) {
        a0 += __shfl_xor(a0, off, 32);
        a1 += __shfl_xor(a1, off, 32);
        a2 += __shfl_xor(a2, off, 32);
    }
    if (lane == 0) {
        a0 += bg[0]; a1 += bg[1]; a2 += bg[2];
        float m = fmaxf(a0, fmaxf(a1, a2));
        float e0 = __expf(a0 - m), e1 = __expf(a1 - m), e2 = __expf(a2 - m);
        float s = e0 + e1 + e2;
        gw[(size_t)n * HOPS + 0] = e0 / s;
        gw[(size_t)n * HOPS + 1] = e1 / s;
        gw[(size_t)n * HOPS + 2] = e2 / s;
    }
}

// ---------------------------------------------------------------------------
// Per-hop scratch init: segmax = -inf, segsum = 0
// ---------------------------------------------------------------------------
__global__ void init_seg_kernel(float* __restrict__ segmax, float* __restrict__ segsum,
                                int count)
{
    int i = blockIdx.x * blockDim.x + threadIdx.x;
    if (i < count) { segmax[i] = -__builtin_inff(); segsum[i] = 0.f; }
}

__device__ inline void atomicMaxF32(float* addr, float val) {
    if (val >= 0.0f) atomicMax((int*)addr, __float_as_int(val));
    else             atomicMin((unsigned int*)addr, __float_as_uint(val));
}

// ---------------------------------------------------------------------------
// Pass 1: per-edge logits + segment max.  One wave per edge; lane covers
// channels c = lane and lane+32 for each of the 4 heads.
// e >= estride encodes self-loop edge (src = dst = e - estride).
// ---------------------------------------------------------------------------
__global__ __launch_bounds__(256)
void edge_logits_kernel(const float* __restrict__ xl, const float* __restrict__ xr,
                        const int* __restrict__ ei, int nedge, int estride,
                        const float* __restrict__ att,   // [H][C] for this hop
                        float* __restrict__ logits,      // [nedge][H]
                        float* __restrict__ segmax)      // [N][H]
{
    int e = blockIdx.x * 8 + (threadIdx.x >> 5);
    if (e >= nedge) return;
    int lane = threadIdx.x & 31;
    int src, dst;
    if (e < estride) { src = ei[e]; dst = ei[estride + e]; }
    else             { src = dst = e - estride; }

    const float* pl = xl + (size_t)src * HC;
    const float* pr = xr + (size_t)dst * HC;
    float lg[H_HEADS];
#pragma unroll
    for (int h = 0; h < H_HEADS; ++h) {
        int i0 = h * C_DIM + lane;
        int i1 = i0 + 32;
        float s0 = pl[i0] + pr[i0];
        s0 = (s0 > 0.f) ? s0 : s0 * NEG_SLOPE;
        float s1 = pl[i1] + pr[i1];
        s1 = (s1 > 0.f) ? s1 : s1 * NEG_SLOPE;
        float p = att[i0] * s0 + att[i1] * s1;
#pragma unroll
        for (int off = 16; off > 0; off >>= 1) p += __shfl_xor(p, off, 32);
        lg[h] = p;
    }
    if (lane == 0) {
#pragma unroll
        for (int h = 0; h < H_HEADS; ++h) {
            logits[(size_t)e * H_HEADS + h] = lg[h];
            atomicMaxF32(&segmax[(size_t)dst * H_HEADS + h], lg[h]);
        }
    }
}

// ---------------------------------------------------------------------------
// Pass 2: ex = exp(logit - segmax[dst]); segsum[dst] += ex.  Thread per (e,h).
// Overwrites logits buffer with ex in place.
// ---------------------------------------------------------------------------
__global__ void edge_softmax_kernel(const int* __restrict__ ei, int nedge, int estride,
                                    const float* __restrict__ segmax,
                                    float* __restrict__ logits_ex,
                                    float* __restrict__ segsum)
{
    int idx = blockIdx.x * blockDim.x + threadIdx.x;
    if (idx >= nedge * H_HEADS) return;
    int e = idx >> 2, h = idx & 3;
    int dst = (e < estride) ? ei[estride + e] : (e - estride);
    float m  = segmax[(size_t)dst * H_HEADS + h];
    float ex = __expf(logits_ex[idx] - m);
    logits_ex[idx] = ex;
    atomicAdd(&segsum[(size_t)dst * H_HEADS + h], ex);
}

// ---------------------------------------------------------------------------
// Pass 3: msg[dst][hop][c] += (1/H) * sum_h xl[src][h*C+c] * alpha[e][h]
// One wave per edge; lane covers c = lane and lane+32.
// ---------------------------------------------------------------------------
__global__ __launch_bounds__(256)
void edge_aggregate_kernel(const float* __restrict__ xl, const int* __restrict__ ei,
                           int nedge, int estride,
                           const float* __restrict__ ex, const float* __restrict__ segsum,
                           float* __restrict__ msg, int hop)
{
    int e = blockIdx.x * 8 + (threadIdx.x >> 5);
    if (e >= nedge) return;
    int lane = threadIdx.x & 31;
    int src, dst;
    if (e < estride) { src = ei[e]; dst = ei[estride + e]; }
    else             { src = dst = e - estride; }

    float w[H_HEADS];
#pragma unroll
    for (int h = 0; h < H_HEADS; ++h)
        w[h] = (1.0f / H_HEADS) * ex[(size_t)e * H_HEADS + h] /
               (segsum[(size_t)dst * H_HEADS + h] + 1e-16f);

    const float* pl = xl + (size_t)src * HC;
    float acc0 = 0.f, acc1 = 0.f;
#pragma unroll
    for (int h = 0; h < H_HEADS; ++h) {
        acc0 += pl[h * C_DIM + lane]      * w[h];
        acc1 += pl[h * C_DIM + lane + 32] * w[h];
    }
    float* mrow = msg + ((size_t)dst * HOPS + hop) * C_DIM;
    atomicAdd(&mrow[lane],      acc0);
    atomicAdd(&mrow[lane + 32], acc1);
}

// ---------------------------------------------------------------------------
// Final: out[n][c] = sum_k (msg[n][k][c] + bias[k][c]) * gatew[n][k]
// ---------------------------------------------------------------------------
__global__ void combine_kernel(const float* __restrict__ msg, const float* __restrict__ bias,
                               const float* __restrict__ gw, float* __restrict__ out,
                               int n_nodes)
{
    int idx = blockIdx.x * blockDim.x + threadIdx.x;
    if (idx >= n_nodes * C_DIM) return;
    int n = idx >> 6, c = idx & 63;
    float acc = 0.f;
#pragma unroll
    for (int k = 0; k < HOPS; ++k)
        acc += (msg[((size_t)n * HOPS + k) * C_DIM + c] + bias[k * C_DIM + c]) *
               gw[(size_t)n * HOPS + k];
    out[idx] = acc;
}

// ---------------------------------------------------------------------------
extern "C" void kernel_launch(void* const* d_in, const int* in_sizes, int n_in,
                              void* d_out, int out_size, void* d_ws, size_t ws_size,
                              hipStream_t stream)
{
    const float* x      = (const float*)d_in[0];
    const int*   ei[3]  = { (const int*)d_in[1], (const int*)d_in[2], (const int*)d_in[3] };
    const float* W_l    = (const float*)d_in[4];
    const float* b_l    = (const float*)d_in[5];
    const float* W_r    = (const float*)d_in[6];
    const float* b_r    = (const float*)d_in[7];
    const float* att    = (const float*)d_in[8];   // [HOPS][H][C]
    const float* bias   = (const float*)d_in[9];   // [HOPS][C]
    const float* W_gate = (const float*)d_in[10];
    const float* b_gate = (const float*)d_in[11];

    const int N = in_sizes[0] / IN_DIM;            // 50000
    const int E = in_sizes[1] / 2;                 // 400000

    // Workspace carve-up (floats)
    float* ws    = (float*)d_ws;
    float* xl    = ws;                 ws += (size_t)N * HC;          // 12.8M
    float* xr    = ws;                 ws += (size_t)N * HC;          // 12.8M
    float* msg   = ws;                 ws += (size_t)N * HOPS * C_DIM;// 9.6M
    float* gatew = ws;                 ws += (size_t)N * HOPS;
    float* segmx = ws;                 ws += (size_t)N * H_HEADS;
    float* segsm = ws;                 ws += (size_t)N * H_HEADS;
    float* lgex  = ws;                 /* (E+N)*H */

    // 1) Shared projections via WMMA (16x32 tile per wave -> grid.y = 2)
    dim3 gg((N + 15) / 16, 2);
    proj_gemm_wmma<<<gg, 128, 0, stream>>>(x, W_l, b_l, xl, N);
    proj_gemm_wmma<<<gg, 128, 0, stream>>>(x, W_r, b_r, xr, N);

    // 2) Gate weights
    gate_kernel<<<(N + 7) / 8, 256, 0, stream>>>(x, W_gate, b_gate, gatew, N);

    // 3) Zero message accumulator
    hipMemsetAsync(msg, 0, (size_t)N * HOPS * C_DIM * sizeof(float), stream);

    // 4) Hops
    for (int k = 0; k < HOPS; ++k) {
        int nedge = (k == 0) ? (E + N) : E;   // hop 0 has self-loops appended
        init_seg_kernel<<<(N * H_HEADS + 255) / 256, 256, 0, stream>>>(segmx, segsm, N * H_HEADS);
        edge_logits_kernel<<<(nedge + 7) / 8, 256, 0, stream>>>(
            xl, xr, ei[k], nedge, E, att + (size_t)k * H_HEADS * C_DIM, lgex, segmx);
        edge_softmax_kernel<<<(nedge * H_HEADS + 255) / 256, 256, 0, stream>>>(
            ei[k], nedge, E, segmx, lgex, segsm);
        edge_aggregate_kernel<<<(nedge + 7) / 8, 256, 0, stream>>>(
            xl, ei[k], nedge, E, lgex, segsm, msg, k);
    }

    // 5) Gated combine with hop biases
    combine_kernel<<<(N * C_DIM + 255) / 256, 256, 0, stream>>>(msg, bias, gatew, (float*)d_out, N);
}